// VectorQuantizer_13451837571400
// MI455X (gfx1250) — compile-verified
//
#include <hip/hip_runtime.h>

#define N_TOTAL 65536
#define D 64
#define V 1024
#define PITCH 66                 // padded floats per codebook row in LDS (even -> 8B aligned b64 reads)
#define WAVES_PER_BLOCK 8
#define ROWS_PER_WAVE 16

typedef float v2f __attribute__((ext_vector_type(2)));
typedef float v8f __attribute__((ext_vector_type(8)));

// LDS layout (single array so the group-segment offset of element 0 is 0,
// which the async-load inline asm relies on for its LDS byte address):
//   [0, V*PITCH)             codebook, row pitch 66 floats
//   [V*PITCH, V*PITCH+V)     per-code squared norms |c|^2
//   [V*PITCH+V, +128)        per-wave token scratch (as ints)
#define LDS_FLOATS (V * PITCH + V + 128)

__global__ __launch_bounds__(256) void vq_wmma_kernel(const float* __restrict__ z,
                                                      const float* __restrict__ cb,
                                                      float* __restrict__ outQ,
                                                      float* __restrict__ outTok,
                                                      float* __restrict__ outCb)
{
    __shared__ __align__(16) float lds[LDS_FLOATS];

    const int tid  = threadIdx.x;
    const int lane = tid & 31;
    const int wave = tid >> 5;

    // ---------------- Stage codebook into LDS via async DMA ----------------
    // 1024 rows x 16 chunks of 16B; 256 threads -> 64 chunks each.
    // Dest is row-padded (264B pitch) so fragment reads below are conflict-light.
    #pragma unroll 4
    for (int i = 0; i < 64; ++i) {
        int ch   = tid + i * 256;
        int row  = ch >> 4;
        int part = ch & 15;
        unsigned gOff = (unsigned)(row * (D * 4) + part * 16);
        unsigned lOff = (unsigned)(row * (PITCH * 4) + part * 16);
        asm volatile("global_load_async_to_lds_b128 %0, %1, %2"
                     :: "v"(lOff), "v"(gOff), "s"(cb) : "memory");
    }
    asm volatile("s_wait_asynccnt 0x0" ::: "memory");
    __syncthreads();

    // ---------------- Codebook squared norms ----------------
    for (int r = tid; r < V; r += 256) {
        const float* row = &lds[r * PITCH];
        float s = 0.0f;
        #pragma unroll
        for (int k = 0; k < D; ++k) s += row[k] * row[k];
        lds[V * PITCH + r] = s;
    }
    __syncthreads();

    // ---------------- Per-wave 16-row tile ----------------
    const int rowBase = (blockIdx.x * WAVES_PER_BLOCK + wave) * ROWS_PER_WAVE;
    const int mrow  = lane & 15;        // A: M row / B: N col / C-D: N col
    const int khalf = (lane >> 4) * 2;  // lanes 16-31 hold K=2,3 of each 4-step

    // Preload A fragments for all K=64 (16 steps of K=4), f32 16x4 layout:
    // lane L: reg0 = z[M][4s+khalf], reg1 = z[M][4s+khalf+1]
    v2f afrag[16];
    const float* zrow = z + (size_t)(rowBase + mrow) * D + khalf;
    #pragma unroll
    for (int s = 0; s < 16; ++s)
        afrag[s] = *(const v2f*)(zrow + s * 4);

    float bestv[8];
    int   besti[8];
    #pragma unroll
    for (int j = 0; j < 8; ++j) { bestv[j] = 3.4e38f; besti[j] = 0; }

    // Sweep 64 column tiles of the codebook
    for (int c = 0; c < 64; ++c) {
        // B fragment (4x16, B[k][n] = cb[n][k]): same per-lane pattern as A,
        // read from LDS (ds_load_b64 pairs, 8B aligned thanks to even PITCH).
        const float* bbase = &lds[(c * 16 + mrow) * PITCH + khalf];

        // 4 independent accumulators -> 4-way WMMA ILP instead of a 16-deep
        // serial RAW chain on a single accumulator (we run only 2 waves/SIMD
        // because LDS caps occupancy, so in-wave ILP is what hides WMMA latency).
        v8f acc0 = {0.f,0.f,0.f,0.f,0.f,0.f,0.f,0.f};
        v8f acc1 = acc0, acc2 = acc0, acc3 = acc0;
        #pragma unroll
        for (int s = 0; s < 16; s += 4) {
            v2f b0 = *(const v2f*)(bbase + (s + 0) * 4);
            v2f b1 = *(const v2f*)(bbase + (s + 1) * 4);
            v2f b2 = *(const v2f*)(bbase + (s + 2) * 4);
            v2f b3 = *(const v2f*)(bbase + (s + 3) * 4);
            acc0 = __builtin_amdgcn_wmma_f32_16x16x4_f32(
                       false, afrag[s + 0], false, b0, (short)0, acc0, false, false);
            acc1 = __builtin_amdgcn_wmma_f32_16x16x4_f32(
                       false, afrag[s + 1], false, b1, (short)0, acc1, false, false);
            acc2 = __builtin_amdgcn_wmma_f32_16x16x4_f32(
                       false, afrag[s + 2], false, b2, (short)0, acc2, false, false);
            acc3 = __builtin_amdgcn_wmma_f32_16x16x4_f32(
                       false, afrag[s + 3], false, b3, (short)0, acc3, false, false);
        }

        // score = |c|^2 - 2*(z.c); |z|^2 dropped (constant per row)
        float ncol = lds[V * PITCH + c * 16 + mrow];
        int   idx  = c * 16 + mrow;
        #pragma unroll
        for (int j = 0; j < 8; ++j) {
            float d3  = (acc0[j] + acc1[j]) + (acc2[j] + acc3[j]);
            float scr = ncol - 2.0f * d3;
            bool better = scr < bestv[j];     // strict < keeps first (lowest) index
            bestv[j] = better ? scr : bestv[j];
            besti[j] = better ? idx : besti[j];
        }
    }

    // Cross-lane min+argmin within each 16-lane half (lane = column class).
    // C/D layout: reg j on lanes 0-15 -> row j; on lanes 16-31 -> row j+8.
    #pragma unroll
    for (int j = 0; j < 8; ++j) {
        #pragma unroll
        for (int off = 8; off >= 1; off >>= 1) {
            float ov = __shfl_xor(bestv[j], off, 32);
            int   oi = __shfl_xor(besti[j], off, 32);
            if (ov < bestv[j] || (ov == bestv[j] && oi < besti[j])) {
                bestv[j] = ov; besti[j] = oi;
            }
        }
    }

    // Lanes 0 and 16 hold winners for rows j and j+8
    int* tokLds = (int*)&lds[V * PITCH + V];
    if ((lane & 15) == 0) {
        int mbase = (lane >> 4) * 8;
        #pragma unroll
        for (int j = 0; j < 8; ++j) {
            int m = mbase + j;
            tokLds[wave * 16 + m] = besti[j];
            outTok[rowBase + m] = (float)besti[j];
        }
    }
    __syncthreads();

    // Gather z_quantized = codebook[token] straight from LDS-resident codebook.
    // Each lane moves 2 floats per row (32 lanes * 2 = 64 = D).
    #pragma unroll
    for (int r = 0; r < ROWS_PER_WAVE; ++r) {
        int tok = tokLds[wave * 16 + r];
        v2f val = *(const v2f*)&lds[tok * PITCH + lane * 2];
        *(v2f*)(outQ + (size_t)(rowBase + r) * D + lane * 2) = val;
    }

    // Codebook passthrough output (third tuple element), block 0 only.
    if (blockIdx.x == 0) {
        for (int i = tid; i < V * D; i += 256) {
            int r = i >> 6, k = i & 63;
            outCb[i] = lds[r * PITCH + k];
        }
    }
}

extern "C" void kernel_launch(void* const* d_in, const int* in_sizes, int n_in,
                              void* d_out, int out_size, void* d_ws, size_t ws_size,
                              hipStream_t stream) {
    const float* z  = (const float*)d_in[0];   // [64,32,32,64] f32
    const float* cb = (const float*)d_in[1];   // [1024,64] f32
    float* out    = (float*)d_out;
    float* outQ   = out;                                   // 65536*64
    float* outTok = out + (size_t)N_TOTAL * D;             // 65536
    float* outCb  = outTok + N_TOTAL;                      // 1024*64

    const int blocks = N_TOTAL / (WAVES_PER_BLOCK * ROWS_PER_WAVE); // 512
    vq_wmma_kernel<<<blocks, 256, 0, stream>>>(z, cb, outQ, outTok, outCb);
}